// AR_Layer_31748398252613
// MI455X (gfx1250) — compile-verified
//
#include <hip/hip_runtime.h>

// AR(2) normalizing-flow layer for MI455X (gfx1250, wave32).
//
// Cholesky of a stationary AR(2) Toeplitz covariance == the AR innovation
// recursion, so z_out = L z is a blocked linear scan: each 16-step block is a
// 16x16 lower-triangular Toeplitz operator (impulse response h, scaled by
// sigma) applied with V_WMMA_F32_16X16X4_F32 in full f32 precision; the
// block-boundary carry enters via phi0[j]=h[j+1], phi1[j]=a1*h[j] and is
// exchanged with ds_bpermute (lanes 15/31 and 14/30 of the previous D tile).
// HBM-bound: ~134 MB traffic -> ~6 us at 23.3 TB/s; compute is negligible,
// so f32 WMMA (not f16) is the right precision choice.

typedef __attribute__((ext_vector_type(2))) float v2f;
typedef __attribute__((ext_vector_type(8))) float v8f;

#define KK 4
#define MM 64
#define DD 64
#define TT 1024
#define KM 256
#define NB (TT / 16)   // 64 time blocks

__device__ __forceinline__ void ar2_chol_params(float a0, float a1, float s,
                                                float& l00, float& l10, float& l11) {
  // gamma0/gamma1 of the stationary AR(2), then the 2x2 Cholesky init.
  float s2 = s * s;
  float g0 = s2 / (1.0f - (a0 * a0 + a1 * a0 * a0) / (1.0f - a1) - a1 * a1);
  float g1 = a0 * g0 / (1.0f - a1);
  l00 = sqrtf(g0);
  l10 = g1 / l00;
  l11 = sqrtf(g0 - l10 * l10);
}

__global__ __launch_bounds__(256) void ar2_wmma_scan_kernel(
    const float* __restrict__ z, const float* __restrict__ alpha,
    const float* __restrict__ log_sigma, float* __restrict__ zout) {
  const int d     = blockIdx.x;           // 0..63
  const int lane  = threadIdx.x & 31;
  const int wv    = threadIdx.x >> 5;     // 0..7
  const int tile  = blockIdx.y * 8 + wv;  // 0..15 -> km tile of 16 rows
  const int jlane = lane & 15;            // N (t offset within block)
  const int hi    = lane >> 4;            // 0 or 1
  const int row0  = tile * 16;            // km base of this tile

  const float a0 = alpha[2 * d + 0];
  const float a1 = alpha[2 * d + 1];
  const float s  = __expf(log_sigma[d]);

  // h / r0 / r1 in LDS so per-lane B construction is plain addressed loads
  // (private variable-index arrays would become v_cndmask ladders).
  __shared__ float hsh[17];
  __shared__ float rsh[2][16];
  if (threadIdx.x == 0) {
    float l00, l10, l11;
    ar2_chol_params(a0, a1, s, l00, l10, l11);
    hsh[0] = 1.0f; hsh[1] = a0;
    for (int i = 2; i < 17; ++i) hsh[i] = a0 * hsh[i - 1] + a1 * hsh[i - 2];
    rsh[0][0] = l00; rsh[0][1] = l10;   // response to e0 (stationary init)
    rsh[1][0] = 0.f; rsh[1][1] = l11;   // response to e1
    for (int i = 2; i < 16; ++i) {
      rsh[0][i] = a0 * rsh[0][i - 1] + a1 * rsh[0][i - 2];
      rsh[1][i] = a0 * rsh[1][i - 1] + a1 * rsh[1][i - 2];
    }
  }
  __syncthreads();

  // Carry coefficients for output position j = jlane.
  const float phi0 = hsh[jlane + 1];
  const float phi1 = a1 * hsh[jlane];

  // B operand (4x16 per K-chunk), f32 layout: lane holds column N=jlane,
  // v2f = { B[k0][N], B[k0+1][N] }, k0 = 4*cc + 2*hi, B[k][j] = Hs[j][k].
  v2f Brest[4], Bfirst[4];
#pragma unroll
  for (int cc = 0; cc < 4; ++cc) {
#pragma unroll
    for (int e = 0; e < 2; ++e) {
      const int k   = 4 * cc + 2 * hi + e;
      const int idx = jlane - k;
      const float vr = (idx >= 0) ? s * hsh[idx] : 0.0f;
      Brest[cc][e]  = vr;
      Bfirst[cc][e] = (k < 2) ? rsh[k][jlane] : vr;   // k<2 only when hi==0
    }
  }

  // A operand: f32 16x4 layout -> lane holds row M = lane%16, K = 2*hi+{0,1};
  // fold the 2*hi into the base pointer.
  const float* zrow = z + ((size_t)(row0 + jlane) * DD + d) * TT + 2 * hi;
  // D store base: row km = row0 + r + 8*hi, t = t0 + jlane.
  float* orow = zout + ((size_t)(row0 + 8 * hi) * DD + d) * TT + jlane;

  const int src1 = 15 + 16 * hi;  // bpermute lane holding y[t_end]   for our half
  const int src2 = 14 + 16 * hi;  // bpermute lane holding y[t_end-1]

  // ---- block 0 (peeled: stationary-init operator, zero carry) ----
  v8f c;
#pragma unroll
  for (int r = 0; r < 8; ++r) c[r] = 0.0f;
#pragma unroll
  for (int cc = 0; cc < 4; ++cc) {
    v2f a;
    a[0] = zrow[4 * cc];
    a[1] = zrow[4 * cc + 1];
    c = __builtin_amdgcn_wmma_f32_16x16x4_f32(
        false, a, false, Bfirst[cc], (short)0, c, false, false);
  }
#pragma unroll
  for (int r = 0; r < 8; ++r) orow[(size_t)r * DD * TT] = c[r];

  // ---- blocks 1..NB-1: steady-state Toeplitz operator + carry ----
  for (int b = 1; b < NB; ++b) {
    const int t0 = 16 * b;
    if (b + 1 < NB) __builtin_prefetch(zrow + t0 + 16, 0, 0);

    // C init = phi0*y[t0-1] + phi1*y[t0-2], pulled from previous D registers.
    v8f cn;
#pragma unroll
    for (int r = 0; r < 8; ++r) {
      const float y1 = __shfl(c[r], src1, 32);
      const float y2 = __shfl(c[r], src2, 32);
      cn[r] = phi0 * y1 + phi1 * y2;
    }

#pragma unroll
    for (int cc = 0; cc < 4; ++cc) {
      v2f a;
      a[0] = zrow[t0 + 4 * cc];
      a[1] = zrow[t0 + 4 * cc + 1];
      cn = __builtin_amdgcn_wmma_f32_16x16x4_f32(
          false, a, false, Brest[cc], (short)0, cn, false, false);
    }

#pragma unroll
    for (int r = 0; r < 8; ++r) orow[(size_t)r * DD * TT + t0] = cn[r];
    c = cn;
  }
}

__global__ __launch_bounds__(256) void ar2_logdet_kernel(
    const float* __restrict__ sldj, const float* __restrict__ alpha,
    const float* __restrict__ log_sigma, float* __restrict__ out2) {
  __shared__ float part[DD];
  __shared__ float tot;
  const int t = threadIdx.x;
  if (t < DD) {
    const float a0 = alpha[2 * t + 0];
    const float a1 = alpha[2 * t + 1];
    const float s  = __expf(log_sigma[t]);
    float l00, l10, l11;
    ar2_chol_params(a0, a1, s, l00, l10, l11);
    // log|diag(L)| = log l00 + log l11 + (T-2) * log sigma
    part[t] = logf(fabsf(l00)) + logf(fabsf(l11)) + (float)(TT - 2) * log_sigma[t];
  }
  __syncthreads();
  if (t == 0) {
    float acc = 0.0f;
    for (int i = 0; i < DD; ++i) acc += part[i];
    tot = acc;
  }
  __syncthreads();
  if (t < KM) out2[t] = sldj[t] + tot;
}

extern "C" void kernel_launch(void* const* d_in, const int* in_sizes, int n_in,
                              void* d_out, int out_size, void* d_ws, size_t ws_size,
                              hipStream_t stream) {
  const float* z     = (const float*)d_in[0];
  const float* sldj  = (const float*)d_in[1];
  const float* alpha = (const float*)d_in[2];  // (D, 2) row-major
  const float* lsig  = (const float*)d_in[3];  // (D,)
  float* zout = (float*)d_out;                     // (K,M,D,T) flat
  float* out2 = zout + (size_t)KK * MM * DD * TT;  // (K,M) flat

  dim3 grid(DD, 2);  // d x tile-group; 8 waves/block, 1 km-tile each
  ar2_wmma_scan_kernel<<<grid, 256, 0, stream>>>(z, alpha, lsig, zout);
  ar2_logdet_kernel<<<1, 256, 0, stream>>>(sldj, alpha, lsig, out2);
}